// HBilinearUpsample_44976897523757
// MI455X (gfx1250) — compile-verified
//
#include <hip/hip_runtime.h>

// Poincare-ball 2x geodesic upsampling for MI455X (gfx1250, wave32).
//
// Midpoint(x,y) reduces to u*x + v*y where (u,v) depend only on the dots
// (x.x, y.y, x.y); tanh(0.5*atanh(t)) = t/(1+sqrt(1-t^2)) removes all
// transcendentals. Neighbor dots are computed as diagonals of 16x16 Gram
// matrices via V_WMMA_F32_16X16X4_F32 (exact fp32). Memory-bound workload:
// ~168 MB traffic -> ~7us at 23.3 TB/s; input (33 MB) lives in the 192 MB L2.

typedef float v2f __attribute__((ext_vector_type(2)));
typedef float v8f __attribute__((ext_vector_type(8)));

#define EPSF  1e-15f
#define MAXA  (1.0f - 1e-5f)

constexpr int Bn = 8, Cn = 64, Hn = 128, Wn = 128;
constexpr int Ho = 256, Wo = 256;
constexpr int WAVES_PER_BLOCK = 4;
constexpr int CSTR = 68;                    // padded channel stride in LDS
constexpr int STAGE_PER_WAVE = 2 * 17 * CSTR; // 2 rows x 17 cols x 68

// Scalar coefficients of geodesic midpoint m = u*x + v*y on the unit-curvature
// Poincare ball, from x2=|x|^2, y2=|y|^2, xy=<x,y>. Mirrors reference clipping.
__device__ __forceinline__ void midU(float x2, float y2, float xy,
                                     float& u, float& v) {
  float A   = 1.0f - 2.0f * xy + y2;
  float Bc  = 1.0f - x2;
  float D   = 1.0f - 2.0f * xy + x2 * y2;
  float Dc  = fmaxf(D, EPSF);
  float iD  = 1.0f / Dc;
  // |w|^2 where w = (-A*x + Bc*y)/Dc
  float wn2 = (A * A * x2 - 2.0f * A * Bc * xy + Bc * Bc * y2) * iD * iD;
  float wn  = sqrtf(fmaxf(wn2, EPSF));
  float t   = fminf(wn, MAXA);
  // gamma = tanh(0.5*atanh(t))/wn  ==  t / ((1+sqrt(1-t^2)) * wn)
  float g   = t / ((1.0f + sqrtf(fmaxf(1.0f - t * t, 0.0f))) * wn);
  float p   = -g * A * iD;   // second = p*x + q*y
  float q   =  g * Bc * iD;
  float s2  = p * p * x2 + 2.0f * p * q * xy + q * q * y2;
  float xs  = p * x2 + q * xy;
  float den = fmaxf(1.0f + 2.0f * xs + x2 * s2, EPSF);
  float id2 = 1.0f / den;
  u = (1.0f + 2.0f * xs + s2 + Bc * p) * id2;
  v = (Bc * q) * id2;
}

// Extract Gram diagonal element (j,j) from a 16x16 f32 WMMA accumulator.
// Layout: VGPR r holds row r (lanes 0-15) and row r+8 (lanes 16-31).
__device__ __forceinline__ float diagx(v8f m, int j, int src) {
  float res = 0.0f;
#pragma unroll
  for (int r = 0; r < 8; ++r) {
    float t = __shfl(m[r], src, 32);
    if ((j & 7) == r) res = t;
  }
  return res;
}

__global__ __launch_bounds__(WAVES_PER_BLOCK * 32)
void hmid_upsample(const float* __restrict__ x, float* __restrict__ out) {
  __shared__ float lds[WAVES_PER_BLOCK * STAGE_PER_WAVE];

  const int lane = threadIdx.x & 31;
  const int wv   = threadIdx.x >> 5;
  float* sm = lds + wv * STAGE_PER_WAVE;

  // One wave = 16 quads: batch b, top row i, columns j0..j0+15.
  const int gw = blockIdx.x * WAVES_PER_BLOCK + wv;   // 0..8191
  const int jt = gw & 7;
  const int i  = (gw >> 3) & (Hn - 1);
  const int b  = gw >> 10;
  const int j0 = jt * 16;
  const int i1 = min(i + 1, Hn - 1);

  // ---- Stage 2 x 17 x 64 tile into LDS (neighbor cols/rows clamped) ----
  const int baseB = b * Cn * Hn * Wn;
#pragma unroll 4
  for (int it = 0; it < (2 * 17 * Cn) / 32; ++it) {   // 68 iterations exactly
    int idx = it * 32 + lane;
    int c   = idx / 34;
    int rem = idx - c * 34;
    int r   = rem / 17;
    int w   = rem - r * 17;
    int row = r ? i1 : i;
    int col = min(j0 + w, Wn - 1);
    float val = x[baseB + (c * Hn + row) * Wn + col];
    sm[(r * 17 + w) * CSTR + c] = val;
  }
  __syncthreads();

  // ---- Gram phase: 10 chains of 16 x V_WMMA_F32_16X16X4_F32 ----
  const int pl = lane & 15;   // pixel / quad index within tile
  const int hi = lane >> 4;   // K-half selector (A/B fragment layout)

  const float* pT  = sm + (pl) * CSTR;            // row i,  col j0+pl
  const float* pTp = sm + (pl + 1) * CSTR;        // row i,  col j0+pl+1
  const float* pB  = sm + (17 + pl) * CSTR;       // row i1, col j0+pl
  const float* pBp = sm + (17 + pl + 1) * CSTR;   // row i1, col j0+pl+1

  v8f m1 = {}, m2 = {}, m3 = {}, m4 = {}, m5 = {};
  v8f m6 = {}, m7 = {}, m8 = {}, m9 = {}, m10 = {};

#pragma unroll
  for (int t = 0; t < 16; ++t) {
    int c0 = 4 * t + 2 * hi;   // lane holds channels (c0, c0+1): k = 2*hi+{0,1}
    v2f fT  = *(const v2f*)(pT  + c0);
    v2f fTp = *(const v2f*)(pTp + c0);
    v2f fB  = *(const v2f*)(pB  + c0);
    v2f fBp = *(const v2f*)(pBp + c0);
    m1  = __builtin_amdgcn_wmma_f32_16x16x4_f32(false, fT,  false, fT,  (short)0, m1,  false, false); // g00
    m2  = __builtin_amdgcn_wmma_f32_16x16x4_f32(false, fTp, false, fTp, (short)0, m2,  false, false); // g11
    m3  = __builtin_amdgcn_wmma_f32_16x16x4_f32(false, fB,  false, fB,  (short)0, m3,  false, false); // g22
    m4  = __builtin_amdgcn_wmma_f32_16x16x4_f32(false, fBp, false, fBp, (short)0, m4,  false, false); // g33
    m5  = __builtin_amdgcn_wmma_f32_16x16x4_f32(false, fT,  false, fTp, (short)0, m5,  false, false); // g01
    m6  = __builtin_amdgcn_wmma_f32_16x16x4_f32(false, fB,  false, fBp, (short)0, m6,  false, false); // g23
    m7  = __builtin_amdgcn_wmma_f32_16x16x4_f32(false, fT,  false, fB,  (short)0, m7,  false, false); // g02
    m8  = __builtin_amdgcn_wmma_f32_16x16x4_f32(false, fTp, false, fBp, (short)0, m8,  false, false); // g13
    m9  = __builtin_amdgcn_wmma_f32_16x16x4_f32(false, fT,  false, fBp, (short)0, m9,  false, false); // g03
    m10 = __builtin_amdgcn_wmma_f32_16x16x4_f32(false, fTp, false, fB,  (short)0, m10, false, false); // g12
  }

  // ---- Diagonal extraction (lanes 16-31 mirror lanes 0-15) ----
  const int j   = pl;
  const int src = (j < 8) ? j : (j + 16);
  float g00 = diagx(m1,  j, src);
  float g11 = diagx(m2,  j, src);
  float g22 = diagx(m3,  j, src);
  float g33 = diagx(m4,  j, src);
  float g01 = diagx(m5,  j, src);
  float g23 = diagx(m6,  j, src);
  float g02 = diagx(m7,  j, src);
  float g13 = diagx(m8,  j, src);
  float g03 = diagx(m9,  j, src);
  float g12 = diagx(m10, j, src);

  // ---- Scalar midpoint coefficients ----
  float uh, vh, uv, vv, ub, vb, uc, vc;
  midU(g00, g11, g01, uh, vh);   // top horizontal midpoint
  midU(g00, g22, g02, uv, vv);   // left vertical midpoint
  midU(g22, g33, g23, ub, vb);   // bottom horizontal midpoint
  float a2 = uh * uh * g00 + 2.0f * uh * vh * g01 + vh * vh * g11;
  float b2 = ub * ub * g22 + 2.0f * ub * vb * g23 + vb * vb * g33;
  float abd = uh * ub * g02 + uh * vb * g03 + vh * ub * g12 + vh * vb * g13;
  midU(a2, b2, abd, uc, vc);     // center = uc*a + vc*b
  float c0w = uc * uh, c1w = uc * vh, c2w = vc * ub, c3w = vc * vb;

  // ---- Output phase: per channel, packed float2 stores (coalesced) ----
  const int colo = 2 * (j0 + pl);
  const int ro0  = 2 * i;
#pragma unroll 4
  for (int cc = 0; cc < Cn; cc += 2) {
    int c = cc + hi;                     // lanes 0-15 even ch, 16-31 odd ch
    float x0 = pT[c];
    float x1 = pTp[c];
    float x2v = pB[c];
    float x3 = pBp[c];
    float oh = uh * x0 + vh * x1;                          // even row, odd col
    float ov = uv * x0 + vv * x2v;                         // odd row, even col
    float oc = c0w * x0 + c1w * x1 + c2w * x2v + c3w * x3; // odd row, odd col
    int ob = ((b * Cn + c) * Ho + ro0) * Wo + colo;
    v2f r0 = {x0, oh};
    v2f r1 = {ov, oc};
    *(v2f*)(out + ob)      = r0;
    *(v2f*)(out + ob + Wo) = r1;
  }
}

extern "C" void kernel_launch(void* const* d_in, const int* in_sizes, int n_in,
                              void* d_out, int out_size, void* d_ws, size_t ws_size,
                              hipStream_t stream) {
  const float* xin = (const float*)d_in[0];
  float* out = (float*)d_out;
  // 8192 wave-tiles (8 batches * 128 rows * 8 col-tiles), 4 waves per block.
  dim3 grid(8192 / WAVES_PER_BLOCK), block(WAVES_PER_BLOCK * 32);
  hipLaunchKernelGGL(hmid_upsample, grid, block, 0, stream, xin, out);
}